// BoundaryHead_contrast_73289321939605
// MI455X (gfx1250) — compile-verified
//
#include <hip/hip_runtime.h>
#include <hip/hip_bf16.h>
#include <math.h>

#define Bsz 32
#define Lsz 8192
#define Dsz 1024
#define BL  (Bsz * Lsz)
#define TOPK 100
#define UNIT 2

typedef __attribute__((ext_vector_type(2))) float v2f;
typedef __attribute__((ext_vector_type(8))) float v8f;

// ---- WMMA helper: D = A(16x4 f32) * B(4x16 f32) + C(16x16 f32) ----
__device__ __forceinline__ v8f wmma_f32_16x16x4(v2f a, v2f b, v8f c) {
#if defined(__gfx1250__)
    // 8 args: (neg_a, A, neg_b, B, c_mod, C, reuse_a, reuse_b)
    return __builtin_amdgcn_wmma_f32_16x16x4_f32(
        false, a, false, b, (short)0, c, false, false);
#else
    (void)a; (void)b;
    return c;
#endif
}

// =====================================================================
// Kernel 1: fused triple GEMV over x (streams x exactly once at HBM rate)
//   center = sigmoid(x.Wc + bc) * (saliency >= 0)
//   window = x.Ww + bw
//   offset = x.Wo + bo
// One wave per 16-row tile; K-loop of 256 WMMA_F32_16X16X4_F32.
// =====================================================================
__global__ __launch_bounds__(256)
void boundary_gemv_kernel(const float* __restrict__ x,
                          const float* __restrict__ sal,
                          const float* __restrict__ Wc, const float* __restrict__ bc,
                          const float* __restrict__ Ww, const float* __restrict__ bw,
                          const float* __restrict__ Wo, const float* __restrict__ bo,
                          float* __restrict__ ctr,
                          float* __restrict__ win,
                          float* __restrict__ off) {
    __shared__ float wlds[3 * Dsz];   // 12 KB: [Wc | Ww | Wo]

    // cooperative weight staging (one-time; reused by all 8 waves' K-loops)
    for (int i = threadIdx.x; i < Dsz; i += blockDim.x) {
        wlds[i]            = Wc[i];
        wlds[Dsz + i]      = Ww[i];
        wlds[2 * Dsz + i]  = Wo[i];
    }
    __syncthreads();

    const int lane  = threadIdx.x & 31;
    const int wave  = threadIdx.x >> 5;
    const int tile  = blockIdx.x * 8 + wave;      // 16384 tiles total
    const int rowBase = tile * 16;

    const int m     = lane & 15;                  // A row / B column index
    const int khalf = (lane >> 4) << 1;           // lanes 0-15 -> K0/K1, 16-31 -> K2/K3
    const float nz  = (m < 3) ? 1.0f : 0.0f;      // only N=0,1,2 carry weights
    const int wsel  = (m < 3) ? m : 0;

    const float* xrow = x + (size_t)(rowBase + m) * Dsz;
    const float* wcol = wlds + wsel * Dsz;

    v8f a0 = {}, a1 = {}, a2 = {}, a3 = {};       // 4 independent accumulators

#define GEMV_STEP(ACC, KB)                                                    \
    {                                                                         \
        const int k0 = (KB) + khalf;                                          \
        v2f av = *(const v2f*)(xrow + k0);                                    \
        v2f bv;                                                               \
        bv.x = wcol[k0] * nz;                                                 \
        bv.y = wcol[k0 + 1] * nz;                                             \
        ACC = wmma_f32_16x16x4(av, bv, ACC);                                  \
    }

#pragma unroll 4
    for (int kb = 0; kb < Dsz; kb += 16) {
        GEMV_STEP(a0, kb + 0)
        GEMV_STEP(a1, kb + 4)
        GEMV_STEP(a2, kb + 8)
        GEMV_STEP(a3, kb + 12)
    }
#undef GEMV_STEP

    v8f c = a0 + a1 + a2 + a3;

    // Epilogue: D[M][N] lives at lane N (N = lane&15), rows M = r (+8 for hi half).
    // Only N = 0 (center), 1 (window), 2 (offset) are meaningful.
    if (m < 3) {
        const int rofs = (lane >> 4) * 8;
        const float bias = (m == 0) ? bc[0] : (m == 1) ? bw[0] : bo[0];
#pragma unroll
        for (int r = 0; r < 8; ++r) {
            const int row = rowBase + rofs + r;
            float v = c[r] + bias;
            if (m == 0) {
                const float s = 1.0f / (1.0f + __expf(-v));
                const float msk = (sal[row] >= 0.0f) ? 1.0f : 0.0f;
                ctr[row] = s * msk;
            } else if (m == 1) {
                win[row] = v;
            } else {
                off[row] = v;
            }
        }
    }
}

// =====================================================================
// Kernel 2: per-batch 5-wide NMS + iterative top-100 + boundary emit.
// One 256-thread block per batch row. cp staged in LDS (32 KB).
// Tie-breaking matches jax.lax.top_k: equal value -> lower index first.
// =====================================================================
__global__ __launch_bounds__(256)
void boundary_topk_kernel(const float* __restrict__ ctr,
                          const float* __restrict__ win,
                          const float* __restrict__ off,
                          float* __restrict__ out) {
    __shared__ float cp[Lsz];        // 32 KB
    __shared__ float rv[256];
    __shared__ int   ri[256];
    __shared__ float topv[TOPK];
    __shared__ int   topi[TOPK];

    const int b    = blockIdx.x;
    const int tid  = threadIdx.x;
    const int base = b * Lsz;

    // NMS: cp[i] = ctr[i] if it is the max of its 5-window (OOB = -inf), else 0
    for (int i = tid; i < Lsz; i += 256) {
        const float ci = ctr[base + i];
        float h = ci;
        if (i >= 2)        h = fmaxf(h, ctr[base + i - 2]);
        if (i >= 1)        h = fmaxf(h, ctr[base + i - 1]);
        if (i + 1 < Lsz)   h = fmaxf(h, ctr[base + i + 1]);
        if (i + 2 < Lsz)   h = fmaxf(h, ctr[base + i + 2]);
        cp[i] = (h == ci) ? ci : 0.0f;
    }
    __syncthreads();

    // iterative argmax x100
    for (int k = 0; k < TOPK; ++k) {
        float bv = -INFINITY;
        int   bi = 0x7fffffff;
        for (int i = tid; i < Lsz; i += 256) {
            const float v = cp[i];
            if (v > bv || (v == bv && i < bi)) { bv = v; bi = i; }
        }
        rv[tid] = bv;
        ri[tid] = bi;
        __syncthreads();
        for (int s = 128; s > 0; s >>= 1) {
            if (tid < s) {
                const float v2 = rv[tid + s];
                const int   i2 = ri[tid + s];
                if (v2 > rv[tid] || (v2 == rv[tid] && i2 < ri[tid])) {
                    rv[tid] = v2; ri[tid] = i2;
                }
            }
            __syncthreads();
        }
        if (tid == 0) {
            topv[k] = rv[0];
            topi[k] = ri[0];
            cp[ri[0]] = -INFINITY;   // remove winner
        }
        __syncthreads();
    }

    // boundary computation for the 100 winners
    if (tid < TOPK) {
        const int   idx   = topi[tid];
        const float score = topv[tid];
        const float o     = off[base + idx];
        const float w     = fmaxf(win[base + idx], 0.0f);
        const float Lm1   = (float)(Lsz - 1);
        float center = fminf(fmaxf((float)idx + o, 0.0f), Lm1);
        float start  = fminf(fmaxf(center - 0.5f * w, 0.0f), Lm1) * (float)UNIT;
        float end    = fminf(fmaxf(center + 0.5f * w, 0.0f), Lm1) * (float)UNIT + (float)UNIT;
        float* p = out + ((size_t)b * TOPK + tid) * 3;
        p[0] = start;
        p[1] = end;
        p[2] = score;
    }
}

extern "C" void kernel_launch(void* const* d_in, const int* in_sizes, int n_in,
                              void* d_out, int out_size, void* d_ws, size_t ws_size,
                              hipStream_t stream) {
    (void)in_sizes; (void)n_in; (void)out_size; (void)ws_size;

    const float* x   = (const float*)d_in[0];
    const float* sal = (const float*)d_in[1];
    const float* Wc  = (const float*)d_in[2];
    const float* bc  = (const float*)d_in[3];
    const float* Ww  = (const float*)d_in[4];
    const float* bw  = (const float*)d_in[5];
    const float* Wo  = (const float*)d_in[6];
    const float* bo  = (const float*)d_in[7];

    float* ws  = (float*)d_ws;
    float* ctr = ws;             // [B*L]
    float* win = ws + BL;        // [B*L]
    float* off = ws + 2 * BL;    // [B*L]
    float* out = (float*)d_out;  // [B, 100, 3]

    // 16384 row-tiles of 16, 8 waves (one tile each) per 256-thread block
    boundary_gemv_kernel<<<dim3(BL / 16 / 8), dim3(256), 0, stream>>>(
        x, sal, Wc, bc, Ww, bw, Wo, bo, ctr, win, off);

    boundary_topk_kernel<<<dim3(Bsz), dim3(256), 0, stream>>>(ctr, win, off, out);
}